// GCN_48670569398724
// MI455X (gfx1250) — compile-verified
//
#include <hip/hip_runtime.h>

typedef __attribute__((ext_vector_type(2))) float v2f;
typedef __attribute__((ext_vector_type(8))) float v8f;

#define NNODES 100000
#define NEDGES 1600000
#define DIM 64
#define NTILES ((NNODES + 15) / 16)   // 6250

// ---------------------------------------------------------------------------
// 1) Degrees via f32 atomics (matches jax segment_sum of ones)
// ---------------------------------------------------------------------------
__global__ void gcn_degree(const int* __restrict__ src, const int* __restrict__ dst,
                           float* __restrict__ degO, float* __restrict__ degI, int E) {
    int t = blockIdx.x * blockDim.x + threadIdx.x;
    if (t < E) {
        atomicAdd(degO + src[t], 1.0f);
        atomicAdd(degI + dst[t], 1.0f);
    }
}

// 2) In-place deg -> rsqrt(max(deg,1))
__global__ void gcn_norm(float* __restrict__ degO, float* __restrict__ degI, int N) {
    int t = blockIdx.x * blockDim.x + threadIdx.x;
    if (t < N) {
        float a = degO[t]; degO[t] = __frsqrt_rn(a < 1.0f ? 1.0f : a);
        float b = degI[t]; degI[t] = __frsqrt_rn(b < 1.0f ? 1.0f : b);
    }
}

// 3) xn[row] = in[row] * ns[row]  (float4 vectorized, N*16 threads)
__global__ void gcn_scale_rows(const float* __restrict__ in, const float* __restrict__ ns,
                               float* __restrict__ out, int N) {
    int t = blockIdx.x * blockDim.x + threadIdx.x;
    if (t < N * (DIM / 4)) {
        int row = t >> 4;
        float s = ns[row];
        float4 v = ((const float4*)in)[t];
        v.x *= s; v.y *= s; v.z *= s; v.w *= s;
        ((float4*)out)[t] = v;
    }
}

// 4) Scatter-add: one wave per edge, each lane owns a float2 of the 64-f row.
//    xn/m are L2-resident (25.6 MB each << 192 MB L2) so the random access
//    pattern stays on-chip.
__global__ void gcn_scatter(const int* __restrict__ src, const int* __restrict__ dst,
                            const float* __restrict__ xn, float* __restrict__ m, int E) {
    int t = blockIdx.x * blockDim.x + threadIdx.x;
    int e = t >> 5;
    if (e >= E) return;
    int lane = t & 31;
    int s = src[e];
    int d = dst[e];
    v2f v = *(const v2f*)(xn + (size_t)s * DIM + lane * 2);
    float* mp = m + (size_t)d * DIM + lane * 2;
    atomicAdd(mp,     v.x);
    atomicAdd(mp + 1, v.y);
}

// ---------------------------------------------------------------------------
// 5) Fused epilogue GEMM:  out = act( (m * nd[:,None]) @ W + b )
//    f32 WMMA 16x16x4, wave32. One wave -> 16 rows x 64 cols (4 C tiles).
//    W (16KB) + b staged in LDS per block.
// ---------------------------------------------------------------------------
__global__ void __launch_bounds__(256)
gcn_gemm_wmma(const float* __restrict__ m, const float* __restrict__ nd,
              const float* __restrict__ W, const float* __restrict__ bias,
              float* __restrict__ out, int nTiles, int doRelu) {
    __shared__ float Wl[DIM * DIM];
    __shared__ float bl[DIM];

    // Stage W row-major into LDS: 1024 float4 / 256 threads = 4 each.
    #pragma unroll
    for (int i = 0; i < 4; ++i) {
        int idx = threadIdx.x + i * 256;
        ((float4*)Wl)[idx] = ((const float4*)W)[idx];
    }
    if (threadIdx.x < DIM) bl[threadIdx.x] = bias[threadIdx.x];
    __syncthreads();

    const int wave = threadIdx.x >> 5;
    const int lane = threadIdx.x & 31;
    const int tile = blockIdx.x * 8 + wave;
    if (tile >= nTiles) return;          // wave-uniform: EXEC stays all-1s for WMMA

    const int mrow16 = lane & 15;        // M index within tile
    const int kHalf  = lane >> 4;        // 0: K pair {0,1}, 1: K pair {2,3}
    const int row    = tile * 16 + mrow16;

    const float* mrow = m + (size_t)row * DIM;
    const float  ndv  = nd[row];

    v8f acc0 = {}, acc1 = {}, acc2 = {}, acc3 = {};

    #pragma unroll
    for (int kk = 0; kk < 16; ++kk) {
        // A fragment (ISA 16x4 f32 layout): lane holds A[M][4kk+2*kHalf+{0,1}],
        // with norm_dst row-scale folded in.
        v2f a = *(const v2f*)(mrow + 4 * kk + 2 * kHalf);
        a *= ndv;

        const int kb = 4 * kk + 2 * kHalf;
        const float* wb = Wl + kb * DIM + mrow16;   // W[kb][n], W[kb+1][n]

        v2f b0; b0.x = wb[0];  b0.y = wb[DIM + 0];
        v2f b1; b1.x = wb[16]; b1.y = wb[DIM + 16];
        v2f b2; b2.x = wb[32]; b2.y = wb[DIM + 32];
        v2f b3; b3.x = wb[48]; b3.y = wb[DIM + 48];

        acc0 = __builtin_amdgcn_wmma_f32_16x16x4_f32(false, a, false, b0, (short)0, acc0, false, false);
        acc1 = __builtin_amdgcn_wmma_f32_16x16x4_f32(false, a, false, b1, (short)0, acc1, false, false);
        acc2 = __builtin_amdgcn_wmma_f32_16x16x4_f32(false, a, false, b2, (short)0, acc2, false, false);
        acc3 = __builtin_amdgcn_wmma_f32_16x16x4_f32(false, a, false, b3, (short)0, acc3, false, false);
    }

    // C/D layout: VGPR v -> row v (lanes 0-15) / row v+8 (lanes 16-31), col = lane&15.
    const int ncol = lane & 15;
    float* orow = out + (size_t)(tile * 16) * DIM;
    #pragma unroll
    for (int v = 0; v < 8; ++v) {
        const int r = v + 8 * kHalf;
        float* op = orow + (size_t)r * DIM;
        float c0 = acc0[v] + bl[ncol];
        float c1 = acc1[v] + bl[16 + ncol];
        float c2 = acc2[v] + bl[32 + ncol];
        float c3 = acc3[v] + bl[48 + ncol];
        if (doRelu) {
            c0 = c0 > 0.0f ? c0 : 0.0f;
            c1 = c1 > 0.0f ? c1 : 0.0f;
            c2 = c2 > 0.0f ? c2 : 0.0f;
            c3 = c3 > 0.0f ? c3 : 0.0f;
        }
        op[ncol]      = c0;
        op[16 + ncol] = c1;
        op[32 + ncol] = c2;
        op[48 + ncol] = c3;
    }
}

// ---------------------------------------------------------------------------
extern "C" void kernel_launch(void* const* d_in, const int* in_sizes, int n_in,
                              void* d_out, int out_size, void* d_ws, size_t ws_size,
                              hipStream_t stream) {
    const float* x  = (const float*)d_in[0];
    const int*   src = (const int*)d_in[1];
    const int*   dst = (const int*)d_in[2];
    const float* W1 = (const float*)d_in[3];
    const float* b1 = (const float*)d_in[4];
    const float* W2 = (const float*)d_in[5];
    const float* b2 = (const float*)d_in[6];
    float* out = (float*)d_out;

    const int N = NNODES, E = NEDGES;

    // Workspace carve-up (256B aligned): ns, nd, xn, m, h1  (~77.6 MB total)
    auto align256 = [](size_t v) { return (v + 255) & ~(size_t)255; };
    char* p = (char*)d_ws;
    float* ns = (float*)p; p += align256((size_t)N * 4);
    float* nd = (float*)p; p += align256((size_t)N * 4);
    float* xn = (float*)p; p += align256((size_t)N * DIM * 4);
    float* m  = (float*)p; p += align256((size_t)N * DIM * 4);
    float* h1 = (float*)p; p += align256((size_t)N * DIM * 4);
    (void)ws_size; (void)in_sizes; (void)n_in; (void)out_size;

    const int TB = 256;

    // Degrees + norms (deg arrays overwritten in place by norms)
    hipMemsetAsync(ns, 0, (size_t)N * 4, stream);
    hipMemsetAsync(nd, 0, (size_t)N * 4, stream);
    gcn_degree<<<(E + TB - 1) / TB, TB, 0, stream>>>(src, dst, ns, nd, E);
    gcn_norm<<<(N + TB - 1) / TB, TB, 0, stream>>>(ns, nd, N);

    const int scaleThreads   = N * (DIM / 4);
    const int scatterThreads = E * 32;
    const int gemmBlocks     = (NTILES + 7) / 8;

    // ---- Layer 1: h1 = relu((segment_sum((x*ns)[src] -> dst) * nd) @ W1 + b1)
    hipMemsetAsync(m, 0, (size_t)N * DIM * 4, stream);
    gcn_scale_rows<<<(scaleThreads + TB - 1) / TB, TB, 0, stream>>>(x, ns, xn, N);
    gcn_scatter<<<(scatterThreads + TB - 1) / TB, TB, 0, stream>>>(src, dst, xn, m, E);
    gcn_gemm_wmma<<<gemmBlocks, TB, 0, stream>>>(m, nd, W1, b1, h1, NTILES, 1);

    // ---- Layer 2: out = (segment_sum((h1*ns)[src] -> dst) * nd) @ W2 + b2
    hipMemsetAsync(m, 0, (size_t)N * DIM * 4, stream);
    gcn_scale_rows<<<(scaleThreads + TB - 1) / TB, TB, 0, stream>>>(h1, ns, xn, N);
    gcn_scatter<<<(scatterThreads + TB - 1) / TB, TB, 0, stream>>>(src, dst, xn, m, E);
    gcn_gemm_wmma<<<gemmBlocks, TB, 0, stream>>>(m, nd, W2, b2, out, NTILES, 0);
}